// CAM_similarity_87677462380852
// MI455X (gfx1250) — compile-verified
//
#include <hip/hip_runtime.h>
#include <math.h>

#define B_  2
#define N1  25
#define N2  30
#define C_  512
#define SP  36
#define SPP 48                  // padded spatial (3 x 16 WMMA tiles)
#define CE  (C_ * SP)           // 18432 elements per patch (real layout)
#define PCE (C_ * SPP)          // 24576 elements per patch (padded interleaved)
#define NPAT (B_ * (N1 + N2))   // 110 patches total
#define TEMP_INV 40.0f          // 1/0.025
#define BN_EPS 1e-5f

typedef __attribute__((ext_vector_type(2))) float v2f;
typedef __attribute__((ext_vector_type(8))) float v8f;

// ---------------------------------------------------------------------------
// Kernel A: per-patch channel-normalization + feats projection g = f_n . W1feat
// One block per patch. First B_*N1 patches are f1, next B_*N2 are f2.
// fn padded-interleaved layout: fn[patch*PCE + (c>>1)*96 + s*2 + (c&1)],
// s in [0,48), zeros for s >= 36  -> WMMA fragments load unguarded v2f.
// g layout: [patch][6]
// ---------------------------------------------------------------------------
__global__ __launch_bounds__(256)
void norm_project_kernel(const float* __restrict__ f1,
                         const float* __restrict__ f2,
                         const float* __restrict__ W1,
                         float* __restrict__ fn,
                         float* __restrict__ g) {
    const int p = blockIdx.x;
    const int tid = threadIdx.x;
    const int NP1 = B_ * N1;
    const float* src = (p < NP1) ? (f1 + (size_t)p * CE)
                                 : (f2 + (size_t)(p - NP1) * CE);
    float* dst = fn + (size_t)p * PCE;

    __shared__ float ssum[SP];
    __shared__ float sscale[SP];
    __shared__ float gsum[6];

    if (tid < SP) ssum[tid] = 0.0f;
    if (tid < 6)  gsum[tid] = 0.0f;
    __syncthreads();

    // sum of squares per spatial column (norm over c)
    for (int idx = tid; idx < CE; idx += 256) {
        float v = src[idx];
        atomicAdd(&ssum[idx % SP], v * v);   // ds_add_f32
    }
    __syncthreads();
    if (tid < SP) sscale[tid] = 1.0f / fmaxf(sqrtf(ssum[tid]), 1e-12f);
    __syncthreads();

    // normalize + scatter to padded layout + accumulate
    // g[k] = sum f_n[c,s] * W1[(36 + c*36 + s), k]
    float acc[6] = {0.f, 0.f, 0.f, 0.f, 0.f, 0.f};
    for (int pidx = tid; pidx < PCE; pidx += 256) {
        const int cpair = pidx / 96;
        const int rem   = pidx - cpair * 96;
        const int s     = rem >> 1;
        const int c     = cpair * 2 + (rem & 1);
        float v = 0.0f;
        if (s < SP) {
            v = src[c * SP + s] * sscale[s];
            const float* wrow = W1 + (size_t)(SP + c * SP + s) * 6;
#pragma unroll
            for (int k = 0; k < 6; ++k) acc[k] += v * wrow[k];
        }
        dst[pidx] = v;
    }
#pragma unroll
    for (int k = 0; k < 6; ++k) atomicAdd(&gsum[k], acc[k]);
    __syncthreads();
    if (tid < 6) g[p * 6 + tid] = gsum[tid];
}

// ---------------------------------------------------------------------------
// Kernel B: one block per (b,i,j) pair. 288 threads = 9 waves.
// Wave w computes 16x16 tile (mi,ni) of a1 = f1n^T * f2n (36x36, K=512)
// using V_WMMA_F32_16X16X4_F32, then attention heads + output writes.
// ---------------------------------------------------------------------------
__global__ __launch_bounds__(288)
void pair_attention_kernel(const float* __restrict__ f1,
                           const float* __restrict__ f2,
                           const float* __restrict__ W1,
                           const float* __restrict__ b1,
                           const float* __restrict__ gamma,
                           const float* __restrict__ beta,
                           const float* __restrict__ rmean,
                           const float* __restrict__ rvar,
                           const float* __restrict__ W2,
                           const float* __restrict__ b2,
                           const float* __restrict__ fn,
                           const float* __restrict__ g,
                           float* __restrict__ out) {
    const int pair = blockIdx.x;
    const int bb = pair / (N1 * N2);
    const int rem = pair % (N1 * N2);
    const int i = rem / N2;
    const int j = rem % N2;

    const float* Af = fn + (size_t)(bb * N1 + i) * PCE;            // f1n patch
    const float* Bf = fn + (size_t)(B_ * N1 + bb * N2 + j) * PCE;  // f2n patch
    const float* g1 = g + (bb * N1 + i) * 6;
    const float* g2 = g + (B_ * N1 + bb * N2 + j) * 6;

    __shared__ float As[48 * 49];     // a1 tile, padded stride 49
    __shared__ float cm[SP], rm[SP];  // column / row means of a1
    __shared__ float h6a[6], h6b[6];
    __shared__ float x36a[SP], x36b[SP];
    __shared__ float att1[SP], att2[SP];

    const int tid  = threadIdx.x;
    const int wave = tid >> 5;
    const int lane = tid & 31;
    const int half = lane >> 4;
    const int ml   = lane & 15;
    const int mi   = wave / 3, ni = wave % 3;
    const int mrow = mi * 16 + ml;   // s index (rows of a1), < 48 (36..47 pad=0)
    const int ncol = ni * 16 + ml;   // t index (cols of a1), < 48

    // fragment base pointers into the padded-interleaved layout:
    // element (c, s) lives at (c>>1)*96 + s*2 + (c&1); per K-step kk the lane
    // needs c = kk + 2*half and c+1, which are the contiguous pair at
    // pair-index (kk>>1)+half.  -> one unconditional v2f (b64) load each.
    const float* Ab = Af + (size_t)mrow * 2 + (size_t)half * 96;
    const float* Bb = Bf + (size_t)ncol * 2 + (size_t)half * 96;

    v8f cacc = {0.f, 0.f, 0.f, 0.f, 0.f, 0.f, 0.f, 0.f};
#pragma unroll 8
    for (int kk = 0; kk < C_; kk += 4) {
        const int kp = (kk >> 1) * 96;         // 2 c-pairs per K-step of 4
        v2f a = *(const v2f*)(Ab + kp);
        v2f b = *(const v2f*)(Bb + kp);
#if defined(__HIP_DEVICE_COMPILE__)
        cacc = __builtin_amdgcn_wmma_f32_16x16x4_f32(
            false, a, false, b, (short)0, cacc, false, false);
#else
        (void)a; (void)b;
#endif
    }
    // C/D layout: VGPR r -> row (r + 8*half), lane%16 -> col
#pragma unroll
    for (int r = 0; r < 8; ++r) {
        As[(mi * 16 + r + 8 * half) * 49 + ncol] = cacc[r];
    }
    __syncthreads();

    // row / column means of a1 (a_mean1 = col means; a_mean2 = row means)
    if (tid < SP) {
        float cs = 0.f, rs = 0.f;
        for (int s = 0; s < SP; ++s) { cs += As[s * 49 + tid]; rs += As[tid * 49 + s]; }
        cm[tid] = cs * (1.0f / SP);
        rm[tid] = rs * (1.0f / SP);
    }
    __syncthreads();

    // first MLP layer (6 outputs) + eval BN + ReLU, for both attention heads
    if (tid < 12) {
        const int k = tid % 6;
        const bool first = tid < 6;
        const float* mean = first ? cm : rm;
        const float* gg   = first ? g2 : g1;
        float x = b1[k] + gg[k];
        for (int t = 0; t < SP; ++t) x += mean[t] * W1[t * 6 + k];
        x = (x - rmean[k]) * rsqrtf(rvar[k] + BN_EPS) * gamma[k] + beta[k];
        x = fmaxf(x, 0.0f);
        (first ? h6a : h6b)[k] = x;
    }
    __syncthreads();

    // second layer: x36 = relu6 @ W2 + b2
    if (tid < 72) {
        const int t = tid % SP;
        const bool first = tid < SP;
        const float* h = first ? h6a : h6b;
        float x = b2[t];
#pragma unroll
        for (int k = 0; k < 6; ++k) x += h[k] * W2[k * SP + t];
        (first ? x36a : x36b)[t] = x;
    }
    __syncthreads();

    // s1[s] = mean_t a1[s,t]*x36a[t] ; s2[t] = mean_s a1[s,t]*x36b[s]
    if (tid < 72) {
        const int s = tid % SP;
        const bool first = tid < SP;
        float acc = 0.f;
        if (first) { for (int t = 0; t < SP; ++t) acc += As[s * 49 + t] * x36a[t]; }
        else       { for (int t = 0; t < SP; ++t) acc += As[t * 49 + s] * x36b[t]; }
        (first ? att1 : att2)[s] = acc * (1.0f / SP);   // store logits temporarily
    }
    __syncthreads();

    // softmax(logits * 40) + 1, serial per head (trivial work)
    if (tid < 2) {
        float* v = (tid == 0) ? att1 : att2;
        float m = -INFINITY;
        for (int s = 0; s < SP; ++s) m = fmaxf(m, v[s] * TEMP_INV);
        float sum = 0.f;
        for (int s = 0; s < SP; ++s) { float e = __expf(v[s] * TEMP_INV - m); v[s] = e; sum += e; }
        float inv = 1.0f / sum;
        for (int s = 0; s < SP; ++s) v[s] = v[s] * inv + 1.0f;
    }
    __syncthreads();

    // outputs (bandwidth-critical, float4 / b128):
    // out1[b,j,i,c,s] = f1[b,i,c,s]*att1[s]; out2[b,j,i,c,t] = f2[b,j,c,t]*att2[t]
    // float4 stride per thread-iteration = 288*4 = 1152 elements, a multiple of
    // 36 -> the 4 spatial indices (and att scalars) are loop-invariant.
    const float4* f1v = (const float4*)(f1 + (size_t)(bb * N1 + i) * CE);
    const float4* f2v = (const float4*)(f2 + (size_t)(bb * N2 + j) * CE);
    float4* o1v = (float4*)(out + (size_t)((bb * N2 + j) * N1 + i) * CE);
    float4* o2v = (float4*)(out + (size_t)(B_ * N2 * N1) * CE
                                + (size_t)((bb * N2 + j) * N1 + i) * CE);
    const int s0 = (tid * 4) % SP;
    const float a1x = att1[s0], a1y = att1[(s0 + 1) % SP],
                a1z = att1[(s0 + 2) % SP], a1w = att1[(s0 + 3) % SP];
    const float a2x = att2[s0], a2y = att2[(s0 + 1) % SP],
                a2z = att2[(s0 + 2) % SP], a2w = att2[(s0 + 3) % SP];
#pragma unroll 4
    for (int it = 0; it < CE / (288 * 4); ++it) {
        const int vi = tid + it * 288;
        float4 v1 = f1v[vi];
        float4 v2 = f2v[vi];
        float4 r1 = make_float4(v1.x * a1x, v1.y * a1y, v1.z * a1z, v1.w * a1w);
        float4 r2 = make_float4(v2.x * a2x, v2.y * a2y, v2.z * a2z, v2.w * a2w);
        o1v[vi] = r1;
        o2v[vi] = r2;
    }
}

extern "C" void kernel_launch(void* const* d_in, const int* in_sizes, int n_in,
                              void* d_out, int out_size, void* d_ws, size_t ws_size,
                              hipStream_t stream) {
    const float* f1    = (const float*)d_in[0];
    const float* f2    = (const float*)d_in[1];
    const float* W1    = (const float*)d_in[2];
    const float* b1    = (const float*)d_in[3];
    const float* gamma = (const float*)d_in[4];
    const float* beta  = (const float*)d_in[5];
    const float* rmean = (const float*)d_in[6];
    const float* rvar  = (const float*)d_in[7];
    const float* W2    = (const float*)d_in[8];
    const float* b2    = (const float*)d_in[9];
    float* out = (float*)d_out;

    float* fn = (float*)d_ws;                       // NPAT * PCE floats (~10.8 MB)
    float* g  = fn + (size_t)NPAT * PCE;            // NPAT * 6 floats

    norm_project_kernel<<<NPAT, 256, 0, stream>>>(f1, f2, W1, fn, g);
    pair_attention_kernel<<<B_ * N1 * N2, 288, 0, stream>>>(
        f1, f2, W1, b1, gamma, beta, rmean, rvar, W2, b2, fn, g, out);
}